// PINN_37074157699724
// MI455X (gfx1250) — compile-verified
//
#include <hip/hip_runtime.h>
#include <math.h>
#include <stdint.h>

// ---------------------------------------------------------------------------
// PINN 3rd-order jet propagation on CDNA5 (gfx1250), f32 WMMA 16x16x4.
// Channels: 0 value | 1..4 d/dx_i | 5..14 d2 (i<=j) | 15..34 d3 (i<=j<=k)
// Data movement: TDM tensor_load_to_lds for the uniform W panel (TENSORcnt),
// global_load_async_to_lds_b128 for per-wave A bands (ASYNCcnt), with
// synchronous fallback if the builtins are unavailable.
// ---------------------------------------------------------------------------

#define NPTS 4096
#define DIM  128
#define NCH  35
#define NLAY 5
#define TWO_PI 6.28318530717958647692f

typedef __attribute__((ext_vector_type(2))) float v2f;
typedef __attribute__((ext_vector_type(8))) float v8f;

#if __has_builtin(__builtin_amdgcn_tensor_load_to_lds) &&                     \
    __has_builtin(__builtin_amdgcn_global_load_async_to_lds_b128) &&          \
    __has_builtin(__builtin_amdgcn_s_wait_asynccnt) &&                        \
    __has_builtin(__builtin_amdgcn_s_wait_tensorcnt)
#define USE_CDNA5_ASYNC 1
typedef __attribute__((ext_vector_type(4))) unsigned int u32x4;
typedef __attribute__((ext_vector_type(8))) int i32x8;
typedef __attribute__((ext_vector_type(4))) int i32x4;
#define AS1 __attribute__((address_space(1)))
#define AS3 __attribute__((address_space(3)))
#endif

// second-order pair list (channel = 5 + p)
__device__ __constant__ int d_PI[10] = {0,0,0,0,1,1,1,2,2,3};
__device__ __constant__ int d_PJ[10] = {0,1,2,3,1,2,3,2,3,3};
// third-order triple list (channel = 15 + t), i<=j<=k
__device__ __constant__ int d_TI[20] = {0,0,0,0,0,0,0,0,0,0,1,1,1,1,1,1,2,2,2,3};
__device__ __constant__ int d_TJ[20] = {0,0,0,0,1,1,1,2,2,3,1,1,1,2,2,3,2,2,3,3};
__device__ __constant__ int d_TK[20] = {0,1,2,3,1,2,3,2,3,3,1,2,3,2,3,3,2,3,3,3};
// (i,j) -> absolute second-order channel
__device__ __constant__ int d_T2[4][4] = {
    {5,6,7,8},{6,9,10,11},{7,10,12,13},{8,11,13,14}};
// sorted (i<=j<=k) -> absolute third-order channel
__device__ __constant__ int d_T3[4][4][4] = {
    {{15,16,17,18},{0,19,20,21},{0,0,22,23},{0,0,0,24}},
    {{ 0, 0, 0, 0},{0,25,26,27},{0,0,28,29},{0,0,0,30}},
    {{ 0, 0, 0, 0},{0, 0, 0, 0},{0,0,31,32},{0,0,0,33}},
    {{ 0, 0, 0, 0},{0, 0, 0, 0},{0,0, 0, 0},{0,0,0,34}}};

__device__ __forceinline__ int ch3(int i, int j, int k) {
    int t;
    if (i > j) { t = i; i = j; j = t; }
    if (j > k) { t = j; j = k; k = t; }
    if (i > j) { t = i; i = j; j = t; }
    return d_T3[i][j][k];
}

__device__ __forceinline__ void sigma_derivs(float a, float& s0, float& s1,
                                             float& s2, float& s3) {
    float sn, cs;
    sincosf(TWO_PI * a, &sn, &cs);
    s0 = 0.5f * sn;
    s1 = 0.5f * TWO_PI * cs;
    s2 = -0.5f * TWO_PI * TWO_PI * sn;
    s3 = -0.5f * TWO_PI * TWO_PI * TWO_PI * cs;
}

__device__ __forceinline__ uint32_t hash_u32(uint32_t x) {
    x ^= x >> 16; x *= 0x7feb352dU;
    x ^= x >> 15; x *= 0x846ca68bU;
    x ^= x >> 16;
    return x;
}
__device__ __forceinline__ float u01(uint32_t h) {
    return (float)(h >> 8) * (1.0f / 16777216.0f);
}

// ---------------------------------------------------------------------------
// Generic C = A(MxK=128) @ W(128 x ncols) GEMM with f32 WMMA 16x16x4.
// bias[col] added to rows < biasRows (value-channel rows in channel-major).
// Block = 128 threads = 4 waves; each wave -> one 16x16 tile; grid covers
// (M/64) x (ncols/16).
// ---------------------------------------------------------------------------
__global__ __launch_bounds__(128) void gemm_wmma(
    const float* __restrict__ A, const float* __restrict__ W,
    const float* __restrict__ bias, float* __restrict__ out,
    int M, int ncols, int biasRows) {
    __shared__ float sW[DIM * 16];        // W panel [k][16]
    __shared__ float sA[4][16 * 132];     // per-wave A band, padded stride 132

    const int tid  = threadIdx.x;
    const int lane = tid & 31;
    const int wave = tid >> 5;
    const int c0   = blockIdx.y * 16;
    const int r0   = (blockIdx.x * 4 + wave) * 16;
    const float* Ab = A + (size_t)r0 * DIM;

#ifdef USE_CDNA5_ASYNC
    // --- TDM: one 2D descriptor DMAs the whole 128x16 W panel into LDS. ---
    if (wave == 0) {
        uint64_t ga = (uint64_t)(uintptr_t)W + (uint64_t)c0 * 4u;
        uint32_t la = (uint32_t)(uintptr_t)(void*)sW;
        u32x4 g0;
        g0.x = 1u;                                  // count=1 (valid), user
        g0.y = la;                                  // lds_addr
        g0.z = (uint32_t)ga;                        // global_addr[31:0]
        g0.w = (uint32_t)((ga >> 32) & 0x01FFFFFFu) // global_addr[56:32]
               | (2u << 30);                        // type=2 (image)
        i32x8 g1;
        g1[0] = (int)(2u << 16);                    // data_size=2 -> 4 bytes
        g1[1] = (int)((uint32_t)ncols << 16);       // tensor_dim0 lo16
        g1[2] = (int)(((uint32_t)ncols >> 16)       // tensor_dim0 hi16
               | (128u << 16));                     // tensor_dim1 lo16 (=128)
        g1[3] = (int)(16u << 16);                   // dim1 hi16=0 | tile_dim0=16
        g1[4] = (int)128u;                          // tile_dim1=128, tile_dim2=0
        g1[5] = (int)(uint32_t)ncols;               // tensor_dim0_stride lo32
        g1[6] = 0;                                  // stride hi16 | dim1_stride lo
        g1[7] = 0;                                  // dim1_stride hi32
        i32x4 z4 = {0, 0, 0, 0};
        i32x8 z8 = {0, 0, 0, 0, 0, 0, 0, 0};
        __builtin_amdgcn_tensor_load_to_lds(g0, g1, z4, z4, z8, 0);
    }
    // --- Per-wave A band (16x128) via async global->LDS, 128b per lane op. ---
    for (int t = lane * 4; t < 16 * DIM; t += 32 * 4) {
        int row = t >> 7, k = t & 127;
        __builtin_amdgcn_global_load_async_to_lds_b128(
            (AS1 i32x4*)(Ab + row * DIM + k),
            (AS3 i32x4*)(&sA[wave][row * 132 + k]), 0, 0);
    }
    __builtin_amdgcn_s_wait_asynccnt(0);
    if (wave == 0) __builtin_amdgcn_s_wait_tensorcnt(0);
#else
    __builtin_prefetch(W, 0, 0);  // global_prefetch_b8 (gfx1250)
    for (int t = tid; t < DIM * 16; t += 128) {
        int k = t >> 4, c = t & 15;
        sW[t] = W[(size_t)k * ncols + c0 + c];
    }
    for (int t = lane; t < 16 * DIM; t += 32) {
        int row = t >> 7, k = t & 127;
        sA[wave][row * 132 + k] = Ab[row * DIM + k];
    }
#endif
    __syncthreads();

    const int half = lane >> 4;   // 0: K pair {k0,k0+1}, 1: {k0+2,k0+3}
    const int mrow = lane & 15;

    v8f acc = {0.f, 0.f, 0.f, 0.f, 0.f, 0.f, 0.f, 0.f};

#pragma unroll
    for (int k0 = 0; k0 < DIM; k0 += 4) {
        const int kk = k0 + 2 * half;
        v2f a, b;
        a.x = sA[wave][mrow * 132 + kk];
        a.y = sA[wave][mrow * 132 + kk + 1];
        b.x = sW[kk * 16 + mrow];
        b.y = sW[(kk + 1) * 16 + mrow];
        acc = __builtin_amdgcn_wmma_f32_16x16x4_f32(
            false, a, false, b, (short)0, acc, false, false);
    }

#pragma unroll
    for (int v = 0; v < 8; ++v) {
        int rg = r0 + v + 8 * half;
        int cg = c0 + mrow;
        float val = acc[v];
        if (rg < biasRows) val += bias[cg];
        out[(size_t)rg * ncols + cg] = val;
    }
}

// ---------------------------------------------------------------------------
// First layer: analytic jets. a = x.Win + b; da/dx_i = Win[i][u]; higher = 0.
// ---------------------------------------------------------------------------
__global__ void init_jets(const float* __restrict__ X,
                          const float* __restrict__ Win,
                          const float* __restrict__ bin,
                          float* __restrict__ Hj) {
    int idx = blockIdx.x * blockDim.x + threadIdx.x;
    if (idx >= NPTS * DIM) return;
    int n = idx >> 7, u = idx & 127;
    float x0 = X[n * 4 + 0], x1 = X[n * 4 + 1],
          x2 = X[n * 4 + 2], x3 = X[n * 4 + 3];
    float wv[4];
    wv[0] = Win[0 * DIM + u]; wv[1] = Win[1 * DIM + u];
    wv[2] = Win[2 * DIM + u]; wv[3] = Win[3 * DIM + u];
    float a = x0 * wv[0] + x1 * wv[1] + x2 * wv[2] + x3 * wv[3] + bin[u];
    float s0, s1, s2, s3;
    sigma_derivs(a, s0, s1, s2, s3);
    const size_t S = (size_t)NPTS * DIM;
    Hj[idx] = s0;
#pragma unroll
    for (int i = 0; i < 4; ++i) Hj[(size_t)(1 + i) * S + idx] = s1 * wv[i];
#pragma unroll
    for (int p = 0; p < 10; ++p)
        Hj[(size_t)(5 + p) * S + idx] = s2 * wv[d_PI[p]] * wv[d_PJ[p]];
#pragma unroll
    for (int t = 0; t < 20; ++t)
        Hj[(size_t)(15 + t) * S + idx] =
            s3 * wv[d_TI[t]] * wv[d_TJ[t]] * wv[d_TK[t]];
}

// ---------------------------------------------------------------------------
// Faa di Bruno mixing: pre-activation jets -> post-activation jets.
// ---------------------------------------------------------------------------
__global__ void mix_jets(const float* __restrict__ Aj, float* __restrict__ Hj) {
    int idx = blockIdx.x * blockDim.x + threadIdx.x;
    if (idx >= NPTS * DIM) return;
    const size_t S = (size_t)NPTS * DIM;
    float a[NCH];
#pragma unroll
    for (int c = 0; c < NCH; ++c) a[c] = Aj[(size_t)c * S + idx];
    float s0, s1, s2, s3;
    sigma_derivs(a[0], s0, s1, s2, s3);
    float g1[4] = {a[1], a[2], a[3], a[4]};
    Hj[idx] = s0;
#pragma unroll
    for (int i = 0; i < 4; ++i) Hj[(size_t)(1 + i) * S + idx] = s1 * g1[i];
#pragma unroll
    for (int p = 0; p < 10; ++p)
        Hj[(size_t)(5 + p) * S + idx] =
            s2 * g1[d_PI[p]] * g1[d_PJ[p]] + s1 * a[5 + p];
#pragma unroll
    for (int t = 0; t < 20; ++t) {
        int i = d_TI[t], j = d_TJ[t], k = d_TK[t];
        Hj[(size_t)(15 + t) * S + idx] =
            s3 * g1[i] * g1[j] * g1[k] +
            s2 * (a[d_T2[i][j]] * g1[k] + a[d_T2[i][k]] * g1[j] +
                  a[d_T2[j][k]] * g1[i]) +
            s1 * a[15 + t];
    }
}

// ---------------------------------------------------------------------------
// Boundary pass helpers (value-only forward).
// ---------------------------------------------------------------------------
__global__ void boundary_init(const float* __restrict__ X,
                              const float* __restrict__ Win,
                              const float* __restrict__ bin,
                              float* __restrict__ Hb) {
    int idx = blockIdx.x * blockDim.x + threadIdx.x;
    if (idx >= NPTS * DIM) return;
    int n = idx >> 7, u = idx & 127;
    float xt = X[n * 4 + 0];
    float bx = u01(hash_u32(0x9E3779B9u ^ (uint32_t)n));
    float by = u01(hash_u32(0x85EBCA6Bu ^ (uint32_t)n));
    float bz = (hash_u32(0xC2B2AE35u ^ (uint32_t)n) & 1u) ? 1.0f : 0.0f;
    float a = xt * Win[0 * DIM + u] + bx * Win[1 * DIM + u] +
              by * Win[2 * DIM + u] + bz * Win[3 * DIM + u] + bin[u];
    Hb[idx] = 0.5f * sinf(TWO_PI * a);
}

__global__ void act_plain(const float* __restrict__ in,
                          float* __restrict__ out, int n) {
    int idx = blockIdx.x * blockDim.x + threadIdx.x;
    if (idx >= n) return;
    out[idx] = 0.5f * sinf(TWO_PI * in[idx]);
}

__global__ void pad_wout(const float* __restrict__ Wout,
                         const float* __restrict__ bout,
                         float* __restrict__ WoutP,
                         float* __restrict__ boutP) {
    int idx = blockIdx.x * blockDim.x + threadIdx.x;
    if (idx < DIM * 16) {
        int u = idx >> 4, o = idx & 15;
        WoutP[idx] = (o < 4) ? Wout[u * 4 + o] : 0.0f;
    } else if (idx < DIM * 16 + 16) {
        int o = idx - DIM * 16;
        boutP[o] = (o < 4) ? bout[o] : 0.0f;
    }
}

__global__ void zero_acc(float* acc) {
    if (threadIdx.x < 16) acc[threadIdx.x] = 0.0f;
}

// ---------------------------------------------------------------------------
// Loss: reads output jets (35 x N x 16) + boundary outputs; reduces 5 terms.
// ---------------------------------------------------------------------------
__global__ void loss_kernel(const float* __restrict__ Yjet,
                            const float* __restrict__ Yb,
                            const float* __restrict__ Yt,
                            float* __restrict__ acc) {
    int n = blockIdx.x * blockDim.x + threadIdx.x;
    float l_data = 0.f, l_conti = 0.f, l_nse = 0.f, l_ee = 0.f, l_tb = 0.f;
    if (n < NPTS) {
        auto Y = [&](int c, int o) -> float {
            return Yjet[((size_t)c * NPTS + n) * 16 + o];
        };
        const float u = Y(0, 0), v = Y(0, 1), w = Y(0, 2);
        float J[4][4];
#pragma unroll
        for (int o = 0; o < 4; ++o)
#pragma unroll
            for (int i = 0; i < 4; ++i) J[o][i] = Y(1 + i, o);
        auto Hn = [&](int o, int i, int j) -> float { return Y(d_T2[i][j], o); };
        auto G = [&](int o, int i, int j, int k) -> float {
            return Y(ch3(i, j, k), o);
        };
        const float u_x = J[0][1], u_y = J[0][2], u_z = J[0][3];
        const float v_x = J[1][1], v_y = J[1][2], v_z = J[1][3];
        const float w_x = J[2][1], w_y = J[2][2], w_z = J[2][3];
        const float T_t = J[3][0], T_x = J[3][1], T_y = J[3][2], T_z = J[3][3];
        const float s = 8.3666002653e-4f;      // sqrt(PR/RA)
        const float kappa = 1.1952286093e-3f;  // sqrt(1/(0.7e6))

        for (int o = 0; o < 3; ++o) {
            float d = Y(0, o) - Yt[n * 4 + o];
            l_data += d * d;
        }
        {
            float c = u_x + v_y + w_z;
            l_conti = c * c;
        }
        float NSE_u = Hn(2,2,0) - Hn(1,3,0)
            + u * (Hn(2,2,1) - Hn(1,3,1)) + v * (Hn(2,2,2) - Hn(1,3,2))
            + w * (Hn(2,2,3) - Hn(1,3,3))
            - (w_y - v_z) * u_x - (u_z - w_x) * u_y - (v_x - u_y) * u_z
            - s * (G(2,2,1,1) - G(1,3,1,1) + G(2,2,2,2) - G(1,3,2,2)
                   + G(2,2,3,3) - G(1,3,3,3))
            - T_y;
        float NSE_v = Hn(0,3,0) - Hn(2,1,0)
            + u * (Hn(0,3,1) - Hn(2,1,1)) + v * (Hn(0,3,2) - Hn(2,1,2))
            + w * (Hn(0,3,3) - Hn(2,1,3))
            - (w_y - v_z) * v_x - (u_z - w_x) * v_y - (v_x - u_y) * v_z
            - s * (G(0,3,1,1) - G(2,1,1,1) + G(0,3,2,2) - G(2,1,2,2)
                   + G(0,3,3,3) - G(2,1,3,3))
            + T_x;
        float NSE_w = Hn(1,1,0) - Hn(0,2,0)
            + u * (Hn(1,1,1) - Hn(0,2,1)) + v * (Hn(1,1,2) - Hn(0,2,2))
            + w * (Hn(1,1,3) - Hn(0,2,3))
            - (w_y - v_z) * w_x - (u_z - w_x) * w_y - (v_x - u_y) * w_z
            - s * (G(1,1,1,1) - G(0,2,1,1) + G(1,1,2,2) - G(0,2,2,2)
                   + G(1,1,3,3) - G(0,2,3,3));
        l_nse = NSE_u * NSE_u + NSE_v * NSE_v + NSE_w * NSE_w;
        float EE = T_t + u * T_x + v * T_y + w * T_z -
                   kappa * (Hn(3,1,1) + Hn(3,2,2) + Hn(3,3,3));
        l_ee = EE * EE;
        float bz = (hash_u32(0xC2B2AE35u ^ (uint32_t)n) & 1u) ? 1.0f : 0.0f;
        float Ttrue = (bz == 0.0f) ? 0.5f : -0.5f;
        float Tb = Yb[n * 16 + 3];
        float d = Ttrue - Tb;
        l_tb = d * d;
    }
    for (int off = 16; off > 0; off >>= 1) {
        l_data  += __shfl_down(l_data, off, 32);
        l_conti += __shfl_down(l_conti, off, 32);
        l_nse   += __shfl_down(l_nse, off, 32);
        l_ee    += __shfl_down(l_ee, off, 32);
        l_tb    += __shfl_down(l_tb, off, 32);
    }
    if ((threadIdx.x & 31) == 0) {
        atomicAdd(&acc[0], l_data);
        atomicAdd(&acc[1], l_conti);
        atomicAdd(&acc[2], l_nse);
        atomicAdd(&acc[3], l_ee);
        atomicAdd(&acc[4], l_tb);
    }
}

__global__ void finalize(const float* __restrict__ acc,
                         float* __restrict__ out) {
    const float invN = 1.0f / (float)NPTS;
    out[0] = acc[0] * (invN / 3.0f) + acc[1] * invN + acc[2] * (invN / 3.0f) +
             acc[3] * invN + acc[4] * invN;
}

// ---------------------------------------------------------------------------
extern "C" void kernel_launch(void* const* d_in, const int* in_sizes, int n_in,
                              void* d_out, int out_size, void* d_ws,
                              size_t ws_size, hipStream_t stream) {
    (void)in_sizes; (void)n_in; (void)out_size; (void)ws_size;
    const float* inputs = (const float*)d_in[0];
    const float* y_true = (const float*)d_in[1];
    const float* W_in   = (const float*)d_in[2];
    const float* b_in   = (const float*)d_in[3];
    const float* W_hid  = (const float*)d_in[4];
    const float* b_hid  = (const float*)d_in[5];
    const float* W_out  = (const float*)d_in[6];
    const float* b_out  = (const float*)d_in[7];

    float* ws = (float*)d_ws;
    const size_t S = (size_t)NCH * NPTS * DIM;  // one jet buffer
    float* jetA  = ws;
    float* jetB  = jetA + S;
    float* Yjet  = jetB + S;                        // NCH*NPTS*16
    float* Hb0   = Yjet + (size_t)NCH * NPTS * 16;
    float* Hb1   = Hb0 + (size_t)NPTS * DIM;
    float* Yb    = Hb1 + (size_t)NPTS * DIM;        // NPTS*16
    float* WoutP = Yb + (size_t)NPTS * 16;          // 128*16
    float* boutP = WoutP + DIM * 16;                // 16
    float* acc   = boutP + 16;                      // 16

    const int Mjet = NCH * NPTS;  // 143360 (multiple of 64)

    zero_acc<<<1, 32, 0, stream>>>(acc);
    pad_wout<<<9, 256, 0, stream>>>(W_out, b_out, WoutP, boutP);
    init_jets<<<(NPTS * DIM) / 256, 256, 0, stream>>>(inputs, W_in, b_in, jetA);

    for (int l = 0; l < NLAY; ++l) {
        gemm_wmma<<<dim3(Mjet / 64, DIM / 16), 128, 0, stream>>>(
            jetA, W_hid + (size_t)l * DIM * DIM, b_hid + l * DIM, jetB,
            Mjet, DIM, NPTS);
        mix_jets<<<(NPTS * DIM) / 256, 256, 0, stream>>>(jetB, jetA);
    }
    gemm_wmma<<<dim3(Mjet / 64, 1), 128, 0, stream>>>(
        jetA, WoutP, boutP, Yjet, Mjet, 16, NPTS);

    // boundary value-only forward
    boundary_init<<<(NPTS * DIM) / 256, 256, 0, stream>>>(inputs, W_in, b_in,
                                                          Hb0);
    for (int l = 0; l < NLAY; ++l) {
        gemm_wmma<<<dim3(NPTS / 64, DIM / 16), 128, 0, stream>>>(
            Hb0, W_hid + (size_t)l * DIM * DIM, b_hid + l * DIM, Hb1,
            NPTS, DIM, NPTS);
        act_plain<<<(NPTS * DIM) / 256, 256, 0, stream>>>(Hb1, Hb0, NPTS * DIM);
    }
    gemm_wmma<<<dim3(NPTS / 64, 1), 128, 0, stream>>>(
        Hb0, WoutP, boutP, Yb, NPTS, 16, NPTS);

    loss_kernel<<<NPTS / 256, 256, 0, stream>>>(Yjet, Yb, y_true, acc);
    finalize<<<1, 1, 0, stream>>>(acc, (float*)d_out);
}